// Sgc_31138512896566
// MI455X (gfx1250) — compile-verified
//
#include <hip/hip_runtime.h>

typedef __attribute__((ext_vector_type(2))) float v2f;
typedef __attribute__((ext_vector_type(8))) float v8f;

#define DIM 128

// --- deg[i] = 1.0 (self loop) ---------------------------------------------
__global__ __launch_bounds__(256) void k_deg_init(float* __restrict__ deg, int n) {
    int i = blockIdx.x * 256 + threadIdx.x;
    if (i < n) deg[i] = 1.0f;
}

// --- deg[dst] += 1 per edge ------------------------------------------------
__global__ __launch_bounds__(256) void k_deg_acc(const long long* __restrict__ ei,
                                                 float* __restrict__ deg, int E) {
    int e = blockIdx.x * 256 + threadIdx.x;
    if (e < E) atomicAdd(&deg[(int)ei[(size_t)E + e]], 1.0f);
}

// --- dinv[i] = rsqrt(deg[i]) (deg >= 1 always) -----------------------------
__global__ __launch_bounds__(256) void k_dinv(const float* __restrict__ deg,
                                              float* __restrict__ dinv, int n) {
    int i = blockIdx.x * 256 + threadIdx.x;
    if (i < n) dinv[i] = rsqrtf(deg[i]);
}

// --- agg[i][:] = x[i][:] * dinv[i]^2  (self-loop message, also zero-inits) -
__global__ __launch_bounds__(256) void k_agg_init(const float* __restrict__ x,
                                                  const float* __restrict__ dinv,
                                                  float* __restrict__ agg, int n) {
    int t = blockIdx.x * 256 + threadIdx.x;           // one float4 per thread
    if (t >= n * 32) return;                          // 128/4 = 32 chunks per node
    int node = t >> 5;
    float s = dinv[node];
    s = s * s;
    float4 v = ((const float4*)x)[t];
    v.x *= s; v.y *= s; v.z *= s; v.w *= s;
    ((float4*)agg)[t] = v;
}

// --- one wave per edge: gather x[src]*norm, 4 f32 atomics per lane ---------
__global__ __launch_bounds__(256) void k_scatter(const float* __restrict__ x,
                                                 const long long* __restrict__ ei,
                                                 const float* __restrict__ dinv,
                                                 float* __restrict__ agg, int E) {
    int gid  = blockIdx.x * 256 + threadIdx.x;
    int e    = gid >> 5;
    int lane = gid & 31;
    if (e >= E) return;
    int s = (int)ei[e];
    int d = (int)ei[(size_t)E + e];
    float nrm = dinv[s] * dinv[d];
    float4 v = *(const float4*)(x + (size_t)s * DIM + lane * 4);
    float* dst = agg + (size_t)d * DIM + lane * 4;
    atomicAdd(dst + 0, v.x * nrm);
    atomicAdd(dst + 1, v.y * nrm);
    atomicAdd(dst + 2, v.z * nrm);
    atomicAdd(dst + 3, v.w * nrm);
}

// --- h = agg @ W^T + b; out = x + relu(h), fp32 WMMA 16x16x4 ---------------
// One wave per 16-row strip; W staged transposed in LDS (Wt[k][n] = W[n][k]).
__global__ __launch_bounds__(256) void k_gemm(const float* __restrict__ agg,
                                              const float* __restrict__ W,
                                              const float* __restrict__ bias,
                                              const float* __restrict__ x,
                                              float* __restrict__ out, int nstrips) {
    __shared__ float Wt[DIM * DIM];                   // 64 KB
    for (int i = threadIdx.x; i < DIM * DIM; i += 256) {
        int n = i >> 7, k = i & 127;
        Wt[k * DIM + n] = W[i];                       // transpose on store
    }
    __syncthreads();

    int wave = threadIdx.x >> 5;
    int lane = threadIdx.x & 31;
    int strip = blockIdx.x * 8 + wave;
    if (strip >= nstrips) return;

    int row0 = strip << 4;
    int mrow = row0 + (lane & 15);                    // A-frag row
    int kb   = (lane >> 4) << 1;                      // K sub-offset: 0 or 2
    int nlo  = lane & 15;                             // B/C-frag column

    v8f acc[8];
#pragma unroll
    for (int nt = 0; nt < 8; ++nt) {
        v8f z = {0.f, 0.f, 0.f, 0.f, 0.f, 0.f, 0.f, 0.f};
        acc[nt] = z;
    }

    const float* arow = agg + (size_t)mrow * DIM;
    for (int k0 = 0; k0 < DIM; k0 += 4) {
        v2f a;
        a.x = arow[k0 + kb];
        a.y = arow[k0 + kb + 1];
#pragma unroll
        for (int nt = 0; nt < 8; ++nt) {
            v2f bf;
            bf.x = Wt[(k0 + kb) * DIM + nt * 16 + nlo];
            bf.y = Wt[(k0 + kb + 1) * DIM + nt * 16 + nlo];
            acc[nt] = __builtin_amdgcn_wmma_f32_16x16x4_f32(
                false, a, false, bf, (short)0, acc[nt], false, false);
        }
    }

    int rhi = (lane >> 4) << 3;                       // +8 rows for upper half-wave
#pragma unroll
    for (int nt = 0; nt < 8; ++nt) {
        int col = nt * 16 + nlo;
        float bv = bias[col];
#pragma unroll
        for (int r = 0; r < 8; ++r) {
            int row = row0 + rhi + r;
            float h = acc[nt][r] + bv;
            size_t idx = (size_t)row * DIM + col;
            out[idx] = x[idx] + fmaxf(h, 0.0f);
        }
    }
}

extern "C" void kernel_launch(void* const* d_in, const int* in_sizes, int n_in,
                              void* d_out, int out_size, void* d_ws, size_t ws_size,
                              hipStream_t stream) {
    const float*     x  = (const float*)d_in[0];
    const long long* ei = (const long long*)d_in[1];   // int64 edge_index [2, E]
    const float*     W  = (const float*)d_in[2];
    const float*     b  = (const float*)d_in[3];
    float*           out = (float*)d_out;

    int n = in_sizes[0] / DIM;       // 50000 nodes
    int E = in_sizes[1] / 2;         // 1.6M edges

    // workspace layout: agg [n*128] | deg [n] | dinv [n]
    float* agg  = (float*)d_ws;
    float* deg  = agg + (size_t)n * DIM;
    float* dinv = deg + n;

    k_deg_init<<<(n + 255) / 256, 256, 0, stream>>>(deg, n);
    k_deg_acc<<<(E + 255) / 256, 256, 0, stream>>>(ei, deg, E);
    k_dinv<<<(n + 255) / 256, 256, 0, stream>>>(deg, dinv, n);
    k_agg_init<<<(n * 32 + 255) / 256, 256, 0, stream>>>(x, dinv, agg, n);

    long long scat_threads = (long long)E * 32;
    k_scatter<<<(unsigned)((scat_threads + 255) / 256), 256, 0, stream>>>(x, ei, dinv, agg, E);

    int nstrips = n / 16;            // 3125 (n divisible by 16)
    k_gemm<<<(nstrips + 7) / 8, 256, 0, stream>>>(agg, W, b, x, out, nstrips);
}